// MultiHeadAttentionQuantum_65481071410499
// MI455X (gfx1250) — compile-verified
//
#include <hip/hip_runtime.h>
#include <stdint.h>

typedef __attribute__((ext_vector_type(2))) float v2f;
typedef __attribute__((ext_vector_type(8))) float v8f;

// ---------------------------------------------------------------------------
// Kernel 1: build the fixed 16x16 complex unitary U = G28 ... G1  (everything
// after the RY encoder), store as Uout[0..255]=Re, Uout[256..511]=Im.
// One wave; thread t owns column t (t<16); gate application is column-local.
// ---------------------------------------------------------------------------
__global__ void qc_prep_kernel(const float* __restrict__ rand_angles,
                               const float* __restrict__ rx_theta,
                               const float* __restrict__ ry_theta,
                               unsigned long long kinds_packed,
                               unsigned long long wires_packed,
                               float* __restrict__ Uout) {
  __shared__ float Ur[16][16];
  __shared__ float Ui[16][16];
  const int t = threadIdx.x;
  if (t < 16) {
    for (int r = 0; r < 16; ++r) { Ur[r][t] = (r == t) ? 1.0f : 0.0f; Ui[r][t] = 0.0f; }
  }
  __syncthreads();
  const float rx = rx_theta[0];
  const float ry = ry_theta[0];
  for (int i = 0; i < 28; ++i) {
    int kind, w;
    float th;
    if (i < 20) {
      kind = (int)((kinds_packed >> (2 * i)) & 3ull);
      w    = (int)((wires_packed >> (2 * i)) & 3ull);
      th   = rand_angles[i];
    } else {
      int j = i - 20;           // per wire: RX(rx_theta) then RY(ry_theta)
      w    = j >> 1;
      kind = (j & 1) ? 1 : 0;   // 0=rx, 1=ry
      th   = (j & 1) ? ry : rx;
    }
    if (t < 16) {
      const int c = t;
      if (kind == 3) {
        // CNOT: control w, target (w+1)%4. wire w <-> state bit (3-w).
        const int cm = 8 >> w;
        const int tm = 8 >> ((w + 1) & 3);
        for (int r = 0; r < 16; ++r) {
          if ((r & cm) && !(r & tm)) {
            const int r2 = r | tm;
            float ar = Ur[r][c], ai = Ui[r][c];
            Ur[r][c] = Ur[r2][c]; Ui[r][c] = Ui[r2][c];
            Ur[r2][c] = ar;       Ui[r2][c] = ai;
          }
        }
      } else {
        const float ch = cosf(0.5f * th);
        const float sh = sinf(0.5f * th);
        float g00r, g00i, g01r, g01i, g10r, g10i, g11r, g11i;
        if (kind == 0) {        // RX: [[c,-is],[-is,c]]
          g00r = ch; g00i = 0.f; g01r = 0.f; g01i = -sh;
          g10r = 0.f; g10i = -sh; g11r = ch; g11i = 0.f;
        } else if (kind == 1) { // RY: [[c,-s],[s,c]]
          g00r = ch; g00i = 0.f; g01r = -sh; g01i = 0.f;
          g10r = sh; g10i = 0.f; g11r = ch; g11i = 0.f;
        } else {                // RZ: diag(e^{-i th/2}, e^{+i th/2})
          g00r = ch; g00i = -sh; g01r = 0.f; g01i = 0.f;
          g10r = 0.f; g10i = 0.f; g11r = ch; g11i = sh;
        }
        const int m = 8 >> w;
        for (int r = 0; r < 16; ++r) {
          if (!(r & m)) {
            const int r1 = r | m;
            const float x0r = Ur[r][c],  x0i = Ui[r][c];
            const float x1r = Ur[r1][c], x1i = Ui[r1][c];
            Ur[r][c]  = g00r * x0r - g00i * x0i + g01r * x1r - g01i * x1i;
            Ui[r][c]  = g00r * x0i + g00i * x0r + g01r * x1i + g01i * x1r;
            Ur[r1][c] = g10r * x0r - g10i * x0i + g11r * x1r - g11i * x1i;
            Ui[r1][c] = g10r * x0i + g10i * x0r + g11r * x1i + g11i * x1r;
          }
        }
      }
    }
  }
  __syncthreads();
  for (int idx = t; idx < 256; idx += 32) {
    Uout[idx]       = Ur[idx >> 4][idx & 15];
    Uout[256 + idx] = Ui[idx >> 4][idx & 15];
  }
}

// ---------------------------------------------------------------------------
// Kernel 2/5: C[M x N] = A[M x K] @ W^T (+bias), full f32 via WMMA 16x16x4.
// One 16x16 C tile per wave; 8 waves per 256-thread block (cols), grid.x rows.
// f32 A 16x4 fragment: lane L holds M=L%16, K = {2*(L/16), 2*(L/16)+1}.
// B 4x16 fragment mirrored: lane L holds N=L%16, same K pair.
// ---------------------------------------------------------------------------
__global__ void gemm16_wmma_kernel(const float* __restrict__ A,
                                   const float* __restrict__ W,
                                   const float* __restrict__ bias,
                                   float* __restrict__ C,
                                   int K, int N) {
  const int wave = threadIdx.x >> 5;
  const int lane = threadIdx.x & 31;
  const int m0 = blockIdx.x * 16;
  const int n0 = (blockIdx.y * 8 + wave) * 16;
  const int mr = lane & 15;
  const int koff = (lane >> 4) << 1;
  const float* __restrict__ Arow = A + (size_t)(m0 + mr) * K;
  const float* __restrict__ Wrow = W + (size_t)(n0 + mr) * K;  // B[k][n] = W[n][k]
  v8f acc = {0.f, 0.f, 0.f, 0.f, 0.f, 0.f, 0.f, 0.f};
  for (int kk = 0; kk < K; kk += 4) {
    v2f a = *(const v2f*)(Arow + kk + koff);
    v2f b = *(const v2f*)(Wrow + kk + koff);
    acc = __builtin_amdgcn_wmma_f32_16x16x4_f32(false, a, false, b, (short)0, acc,
                                                false, false);
  }
  const int col = n0 + (lane & 15);
  const float bv = bias ? bias[col] : 0.0f;
  const int rbase = m0 + ((lane >> 4) << 3);
#pragma unroll
  for (int i = 0; i < 8; ++i) {
    C[(size_t)(rbase + i) * N + col] = acc[i] + bv;
  }
}

// ---------------------------------------------------------------------------
// Kernel 3: quantum transform. One thread per 4-vector (1.57M rows).
// psi_a = prod_w (bit ? sin : cos)(v_w/2);  phi = U psi;  z_w = sum_s ±|phi_s|^2
// ---------------------------------------------------------------------------
__global__ void quantum_apply_kernel(const float* __restrict__ qkvbuf,
                                     const float* __restrict__ Uw,
                                     float* __restrict__ outQ,
                                     float* __restrict__ outK,
                                     float* __restrict__ outV) {
  __shared__ float Ur[256];
  __shared__ float Ui[256];
  for (int i = threadIdx.x; i < 256; i += blockDim.x) {
    Ur[i] = Uw[i];
    Ui[i] = Uw[256 + i];
  }
  __syncthreads();
  const int g = blockIdx.x * blockDim.x + threadIdx.x;  // < 3*8192*64
  const int s = g / (8192 * 64);
  const int rem = g - s * (8192 * 64);
  const int n = rem >> 6;
  const int h = rem & 63;
  const float* __restrict__ in = qkvbuf + (size_t)n * 768 + s * 256 + h * 4;
  float c0 = cosf(0.5f * in[0]), s0 = sinf(0.5f * in[0]);
  float c1 = cosf(0.5f * in[1]), s1 = sinf(0.5f * in[1]);
  float c2 = cosf(0.5f * in[2]), s2 = sinf(0.5f * in[2]);
  float c3 = cosf(0.5f * in[3]), s3 = sinf(0.5f * in[3]);
  float p01[4] = {c0 * c1, c0 * s1, s0 * c1, s0 * s1};
  float p23[4] = {c2 * c3, c2 * s3, s2 * c3, s2 * s3};
  float psi[16];
#pragma unroll
  for (int a = 0; a < 16; ++a) psi[a] = p01[a >> 2] * p23[a & 3];
  float z0 = 0.f, z1 = 0.f, z2 = 0.f, z3 = 0.f;
#pragma unroll
  for (int sr = 0; sr < 16; ++sr) {
    float fr = 0.f, fi = 0.f;
#pragma unroll
    for (int a = 0; a < 16; ++a) {
      fr += Ur[sr * 16 + a] * psi[a];
      fi += Ui[sr * 16 + a] * psi[a];
    }
    const float p = fr * fr + fi * fi;
    z0 += (sr & 8) ? -p : p;
    z1 += (sr & 4) ? -p : p;
    z2 += (sr & 2) ? -p : p;
    z3 += (sr & 1) ? -p : p;
  }
  float* __restrict__ out =
      ((s == 0) ? outQ : (s == 1) ? outK : outV) + (size_t)n * 256 + h * 4;
  out[0] = z0; out[1] = z1; out[2] = z2; out[3] = z3;
}

// ---------------------------------------------------------------------------
// Kernel 4: per-(b,time) attention slice. S = Q K^T * 0.5 via WMMA (K=4, one
// wmma per 16x16 tile, 4 waves x 4 tiles), softmax over 64, O = P V.
// Output written at the reference's faithful flat layout b*524288+head*8192+t*4.
// ---------------------------------------------------------------------------
__global__ void attn_kernel(const float* __restrict__ Q,
                            const float* __restrict__ Km,
                            const float* __restrict__ V,
                            float* __restrict__ O) {
  __shared__ float S[64][65];
  __shared__ float Vs[64][4];
  const int slice = blockIdx.x;  // b*2048 + time
  const float* __restrict__ Qs = Q + (size_t)slice * 256;
  const float* __restrict__ Ks = Km + (size_t)slice * 256;
  const float* __restrict__ Vg = V + (size_t)slice * 256;
  const int tid = threadIdx.x;  // 128 threads
  for (int i = tid; i < 256; i += 128) Vs[i >> 2][i & 3] = Vg[i];
  const int wave = tid >> 5;
  const int lane = tid & 31;
  const int mr = lane & 15;
  const int koff = (lane >> 4) << 1;
  const int mi = wave;
  v2f a = *(const v2f*)(Qs + (mi * 16 + mr) * 4 + koff);
#pragma unroll
  for (int ni = 0; ni < 4; ++ni) {
    v2f b = *(const v2f*)(Ks + (ni * 16 + mr) * 4 + koff);
    v8f acc = {0.f, 0.f, 0.f, 0.f, 0.f, 0.f, 0.f, 0.f};
    acc = __builtin_amdgcn_wmma_f32_16x16x4_f32(false, a, false, b, (short)0, acc,
                                                false, false);
    const int col = ni * 16 + (lane & 15);
    const int rb = mi * 16 + ((lane >> 4) << 3);
#pragma unroll
    for (int i = 0; i < 8; ++i) S[rb + i][col] = acc[i] * 0.5f;
  }
  __syncthreads();
  if (tid < 64) {
    const int r = tid;
    float mx = -3.4e38f;
    for (int c = 0; c < 64; ++c) mx = fmaxf(mx, S[r][c]);
    float sum = 0.f, o0 = 0.f, o1 = 0.f, o2 = 0.f, o3 = 0.f;
    for (int c = 0; c < 64; ++c) {
      const float e = __expf(S[r][c] - mx);
      sum += e;
      o0 += e * Vs[c][0];
      o1 += e * Vs[c][1];
      o2 += e * Vs[c][2];
      o3 += e * Vs[c][3];
    }
    const float inv = 1.0f / sum;
    const int b = slice >> 11;
    const int time = slice & 2047;
    float* __restrict__ op =
        O + (size_t)b * 524288 + (size_t)r * 8192 + (size_t)time * 4;
    op[0] = o0 * inv; op[1] = o1 * inv; op[2] = o2 * inv; op[3] = o3 * inv;
  }
}

// ---------------------------------------------------------------------------
// Host: replicate np.random.default_rng(1234) -> 20x (integers(4), integers(4))
// SeedSequence pool hashing + PCG64 XSL-RR 128/64 + buffered 32-bit Lemire.
// ---------------------------------------------------------------------------
static void compute_ops_host(uint32_t kinds[20], uint32_t wires[20]) {
  const uint32_t MULT_A = 0x931e8875u, MULT_B = 0x58f38dedu;
  const uint32_t MIX_L = 0xca01f9ddu, MIX_R = 0x4973f715u;
  uint32_t hc = 0x43b0d7e5u;  // INIT_A
  uint32_t pool[4];
  uint32_t entropy0 = 1234u;
  for (int i = 0; i < 4; ++i) {
    uint32_t v = (i < 1) ? entropy0 : 0u;
    v ^= hc; hc *= MULT_A; v *= hc; v ^= v >> 16;
    pool[i] = v;
  }
  for (int sI = 0; sI < 4; ++sI) {
    for (int dI = 0; dI < 4; ++dI) {
      if (sI == dI) continue;
      uint32_t v = pool[sI];
      v ^= hc; hc *= MULT_A; v *= hc; v ^= v >> 16;
      uint32_t r = pool[dI] * MIX_L - v * MIX_R;
      r ^= r >> 16;
      pool[dI] = r;
    }
  }
  uint32_t st[8];
  uint32_t hb = 0x8b51f9ddu;  // INIT_B
  for (int i = 0; i < 8; ++i) {
    uint32_t dv = pool[i & 3];
    dv ^= hb; hb *= MULT_B; dv *= hb; dv ^= dv >> 16;
    st[i] = dv;
  }
  uint64_t w64[4];
  for (int i = 0; i < 4; ++i)
    w64[i] = (uint64_t)st[2 * i] | ((uint64_t)st[2 * i + 1] << 32);
  typedef unsigned __int128 u128;
  const u128 MULT = ((u128)0x2360ed051fc65da4ULL << 64) | 0x4385df649fccf645ULL;
  u128 state = 0;
  u128 inc = ((((u128)w64[2] << 64) | w64[3]) << 1) | 1;
  state = state * MULT + inc;                       // srandom step 1
  state += ((u128)w64[0] << 64) | w64[1];           // += initstate
  state = state * MULT + inc;                       // srandom step 2
  bool has32 = false;
  uint32_t buf32 = 0;
  for (int i = 0; i < 40; ++i) {
    uint32_t u;
    if (has32) {
      has32 = false;
      u = buf32;
    } else {
      state = state * MULT + inc;
      uint64_t hi = (uint64_t)(state >> 64), lo = (uint64_t)state;
      unsigned rot = (unsigned)(state >> 122) & 63u;
      uint64_t x = hi ^ lo;
      uint64_t out = (x >> rot) | (x << ((64u - rot) & 63u));
      buf32 = (uint32_t)(out >> 32);
      has32 = true;
      u = (uint32_t)out;
    }
    uint32_t val = (uint32_t)(((uint64_t)u * 4ull) >> 32);  // Lemire, range 4
    if (i & 1) wires[i >> 1] = val; else kinds[i >> 1] = val;
  }
}

extern "C" void kernel_launch(void* const* d_in, const int* in_sizes, int n_in,
                              void* d_out, int out_size, void* d_ws, size_t ws_size,
                              hipStream_t stream) {
  (void)in_sizes; (void)n_in; (void)out_size; (void)ws_size;
  const float* x           = (const float*)d_in[0];  // (4,2048,256)
  const float* w_qkv       = (const float*)d_in[1];  // (768,256)
  const float* w_out       = (const float*)d_in[2];  // (256,256)
  const float* b_out       = (const float*)d_in[3];  // (256,)
  const float* rx_theta    = (const float*)d_in[4];  // scalar
  const float* ry_theta    = (const float*)d_in[5];  // scalar
  const float* rand_angles = (const float*)d_in[6];  // (20,)
  float* out = (float*)d_out;                        // (4,2048,256) f32

  float* ws      = (float*)d_ws;
  float* U       = ws;                 //  512 floats (pad to 1024)
  float* qkvbuf  = ws + 1024;          //  8192*768
  float* qbuf    = qkvbuf + 6291456;   //  8192*256
  float* kbuf    = qbuf + 2097152;     //  8192*256
  float* vbuf    = kbuf + 2097152;     //  8192*256
  float* attnbuf = qkvbuf;             //  reuse (8192*256), qkvbuf dead by then

  uint32_t kinds[20], wires[20];
  compute_ops_host(kinds, wires);
  unsigned long long kp = 0, wp = 0;
  for (int i = 0; i < 20; ++i) {
    kp |= ((unsigned long long)(kinds[i] & 3u)) << (2 * i);
    wp |= ((unsigned long long)(wires[i] & 3u)) << (2 * i);
  }

  qc_prep_kernel<<<1, 32, 0, stream>>>(rand_angles, rx_theta, ry_theta, kp, wp, U);
  gemm16_wmma_kernel<<<dim3(512, 6), 256, 0, stream>>>(x, w_qkv, nullptr, qkvbuf,
                                                       256, 768);
  quantum_apply_kernel<<<6144, 256, 0, stream>>>(qkvbuf, U, qbuf, kbuf, vbuf);
  attn_kernel<<<8192, 128, 0, stream>>>(qbuf, kbuf, vbuf, attnbuf);
  gemm16_wmma_kernel<<<dim3(512, 2), 256, 0, stream>>>(attnbuf, w_out, b_out, out,
                                                       256, 256);
}